// MixtureGaussianSequenceLabeling_16131897164057
// MI455X (gfx1250) — compile-verified
//
#include <hip/hip_runtime.h>
#include <hip/hip_bf16.h>
#include <math.h>

// ---------------------------------------------------------------------------
// MixtureGaussianSequenceLabeling on MI455X (gfx1250, wave32).
// fp32 WMMA (v_wmma_f32_16x16x4_f32) for all dense 64x64 / 128x128 matmuls;
// cooperative Cholesky + triangular inverse in LDS for SPD inversions.
// All contiguous global tile traffic uses float4 (B128) transactions.
// ---------------------------------------------------------------------------

#define DIMV    64
#define TDV     128
#define ICOMPV  2
#define TCOMPV  4
#define OCOMPV  2
#define MAXKV   4
#define NLABV   50
#define BATCHV  32
#define MAXLENV 128
#define LOG2PI_F 1.8378770664093453f
#define LDP 65   // padded LDS leading dim: stride 65 mod 64 banks = 1 (conflict-free columns)

typedef float v2f __attribute__((ext_vector_type(2)));
typedef float v8f __attribute__((ext_vector_type(8)));

// ---- workspace layout (float offsets) -------------------------------------
enum : size_t {
  OFF_TVAR  = 0,                       // 4*128*128
  OFF_INITS = OFF_TVAR  + 65536,       // 4
  OFF_INITM = OFF_INITS + 4,           // 4*64
  OFF_INITV = OFF_INITM + 256,         // 4*4096
  OFF_CS    = OFF_INITV + 16384,       // 32*4
  OFF_CM    = OFF_CS    + 128,         // 32*4*64
  OFF_CV    = OFF_CM    + 8192,        // 32*4*4096
  OFF_MS    = OFF_CV    + 524288,      // 32*8
  OFF_MM    = OFF_MS    + 256,         // 32*8*64
  OFF_MV    = OFF_MM    + 16384,       // 32*8*4096
  OFF_QS    = OFF_MV    + 1048576,     // 32*32
  OFF_QM    = OFF_QS    + 1024,        // 32*32*64
  OFF_QV    = OFF_QM    + 65536,       // 32*32*4096
  OFF_FS    = OFF_QV    + 4194304,     // 32*4
  OFF_FM    = OFF_FS    + 128,         // 32*4*64
  OFF_FV    = OFF_FM    + 8192,        // 32*4*4096
  OFF_SC8   = OFF_FV    + 524288,      // 32*50*8
  OFF_END   = OFF_SC8   + 12800
};

// ---- WMMA f32 16x16x4 tiled matmul: C(MxN) = A(MxK) * B(KxN) ---------------
__device__ __forceinline__ float mm_elem(const float* P, int ld, int r, int c,
                                         bool tr) {
  return tr ? P[c * ld + r] : P[r * ld + c];
}

__device__ __forceinline__ void wmma_mm(int M, int N, int Kd,
                                        const float* A, int lda, bool ta,
                                        const float* B, int ldb, bool tb,
                                        float* C, int ldc) {
  const int lane   = threadIdx.x & 31;
  const int wave   = threadIdx.x >> 5;
  const int nwaves = blockDim.x >> 5;
  const int half   = lane >> 4;   // 0: K pair (0,1); 1: K pair (2,3)
  const int l      = lane & 15;
  const int ntc    = N >> 4;
  const int ntiles = (M >> 4) * ntc;
  // A fragment pair (k0,k0+1) is contiguous & 8B aligned when !ta and lda even
  const bool veca = (!ta) && ((lda & 1) == 0);
  for (int tile = wave; tile < ntiles; tile += nwaves) {
    const int row0 = (tile / ntc) << 4;
    const int col0 = (tile % ntc) << 4;
    v8f acc = {};
    for (int kb = 0; kb < Kd; kb += 4) {
      const int k0 = kb + half * 2;
      const int ar = row0 + l;
      v2f a, b;
      if (veca) {
        a = *(const v2f*)(A + ar * lda + k0);     // global_load_b64
      } else {
        a.x = mm_elem(A, lda, ar, k0,     ta);
        a.y = mm_elem(A, lda, ar, k0 + 1, ta);
      }
      b.x = mm_elem(B, ldb, k0,     col0 + l, tb);
      b.y = mm_elem(B, ldb, k0 + 1, col0 + l, tb);
      acc = __builtin_amdgcn_wmma_f32_16x16x4_f32(
          false, a, false, b, (short)0, acc, false, false);
    }
#pragma unroll
    for (int r = 0; r < 8; ++r)
      C[(row0 + r + half * 8) * ldc + (col0 + l)] = acc[r];
  }
}

// ---- cooperative 64x64 SPD Cholesky (in place, lower) ----------------------
__device__ void chol64(float* A) {
  const int tid = threadIdx.x;
  for (int k = 0; k < 64; ++k) {
    if (tid == 0) A[k * LDP + k] = sqrtf(A[k * LDP + k]);
    __syncthreads();
    if (tid > k && tid < 64) A[tid * LDP + k] /= A[k * LDP + k];
    __syncthreads();
    if (tid > k && tid < 64) {
      const float lik = A[tid * LDP + k];
      for (int j = k + 1; j <= tid; ++j)
        A[tid * LDP + j] -= lik * A[j * LDP + k];
    }
    __syncthreads();
  }
}

// ---- triangular inverse of lower L -> X (X upper part zeroed) --------------
__device__ void trinv64(const float* L, float* X) {
  const int c = threadIdx.x;
  if (c < 64) {
    for (int i = 0; i < c; ++i) X[i * LDP + c] = 0.f;
    X[c * LDP + c] = 1.f / L[c * LDP + c];
    for (int i = c + 1; i < 64; ++i) {
      float s = 0.f;
      for (int j = c; j < i; ++j) s += L[i * LDP + j] * X[j * LDP + c];
      X[i * LDP + c] = -s / L[i * LDP + i];
    }
  }
  __syncthreads();
}

__device__ __forceinline__ float dot64(const float* a, const float* b) {
  float s = 0.f;
  for (int j = 0; j < 64; ++j) s += a[j] * b[j];
  return s;
}

// ---------------------------------------------------------------------------
// t_var[c] = trans_cho[c]^T @ trans_cho[c]   (128x128, einsum ...ki,...kj)
// ---------------------------------------------------------------------------
__global__ void k_tvar(const float* __restrict__ cho, float* __restrict__ tv) {
  const int c = blockIdx.x;
  const float* A = cho + c * (TDV * TDV);
  wmma_mm(TDV, TDV, TDV, A, TDV, true, A, TDV, false, tv + c * TDV * TDV, TDV);
}

// ---------------------------------------------------------------------------
// initial gaussian_multi_integral_fwd(t_mu, 0, t_var, I) per component
// ---------------------------------------------------------------------------
__global__ void k_init(const float* __restrict__ tmu,
                       const float* __restrict__ tv,
                       float* __restrict__ is, float* __restrict__ im,
                       float* __restrict__ iv) {
  __shared__ float sA[64 * LDP];
  __shared__ float sB[64 * LDP];
  __shared__ float sC[64 * LDP];
  __shared__ float diff[64];
  __shared__ float red[64];
  __shared__ float sh[2];
  const int tid = threadIdx.x;
  const int c = blockIdx.x;
  const float* tvc = tv + c * TDV * TDV;

  if (tid < 64) diff[tid] = -tmu[c * TDV + tid];
  for (int idx = tid; idx < 1024; idx += blockDim.x) {   // float4 tile fill
    const int i = idx >> 4, j0 = (idx & 15) << 2;
    const float4 vv = *(const float4*)(tvc + i * TDV + j0);
    float* d = &sA[i * LDP + j0];
    d[0] = vv.x + (i == j0     ? 1.f : 0.f);
    d[1] = vv.y + (i == j0 + 1 ? 1.f : 0.f);
    d[2] = vv.z + (i == j0 + 2 ? 1.f : 0.f);
    d[3] = vv.w + (i == j0 + 3 ? 1.f : 0.f);
  }
  __syncthreads();
  chol64(sA);
  if (tid == 0) {
    float s = 0.f;
    for (int k = 0; k < 64; ++k) s += logf(sA[k * LDP + k]);
    sh[0] = 2.f * s;
  }
  __syncthreads();
  trinv64(sA, sB);
  wmma_mm(64, 64, 64, sB, LDP, true, sB, LDP, false, sC, LDP);  // Sinv
  __syncthreads();
  wmma_mm(64, 64, 64, tvc + 64 * TDV, TDV, false, sC, LDP, false, sB, LDP); // K
  __syncthreads();
  if (tid < 64)
    im[c * 64 + tid] = tmu[c * TDV + 64 + tid] + dot64(sB + tid * LDP, diff);
  __syncthreads();
  wmma_mm(64, 64, 64, sB, LDP, false, tvc + 64, TDV, false, sA, LDP); // K@Sxy
  __syncthreads();
  for (int idx = tid; idx < 1024; idx += blockDim.x) {
    const int i = idx >> 4, j0 = (idx & 15) << 2;
    const float4 sy = *(const float4*)(tvc + (64 + i) * TDV + 64 + j0);
    const float* m2 = &sA[i * LDP + j0];
    float4 o;
    o.x = sy.x - m2[0]; o.y = sy.y - m2[1];
    o.z = sy.z - m2[2]; o.w = sy.w - m2[3];
    ((float4*)(iv + c * 4096))[idx] = o;
  }
  if (tid < 64) red[tid] = diff[tid] * dot64(sC + tid * LDP, diff);
  __syncthreads();
  if (tid == 0) {
    float m = 0.f;
    for (int i = 0; i < 64; ++i) m += red[i];
    is[c] = -0.5f * (64.f * LOG2PI_F + sh[0] + m);
  }
}

// scatter init components (sorted by score desc) into per-batch carry
__global__ void k_init_scatter(const float* __restrict__ is,
                               const float* __restrict__ im,
                               const float* __restrict__ iv,
                               float* __restrict__ cs, float* __restrict__ cm,
                               float* __restrict__ cv) {
  __shared__ int ssel;
  __shared__ float ssc;
  const int tid = threadIdx.x;
  const int b = blockIdx.x >> 2, k = blockIdx.x & 3;
  if (tid == 0) {
    float s[TCOMPV];
    for (int j = 0; j < TCOMPV; ++j) s[j] = is[j];
    int bi = 0;
    for (int r = 0; r <= k; ++r) {
      float best = -INFINITY; bi = 0;
      for (int j = 0; j < TCOMPV; ++j)
        if (s[j] > best) { best = s[j]; bi = j; }
      if (r < k) s[bi] = -INFINITY; else ssc = best;
    }
    ssel = bi;
  }
  __syncthreads();
  const int j = ssel;
  if (tid == 0) cs[b * 4 + k] = ssc;
  if (tid < 64) cm[(b * 4 + k) * 64 + tid] = im[j * 64 + tid];
  float4* dst = (float4*)(cv + ((size_t)(b * 4 + k) << 12));
  const float4* src = (const float4*)(iv + ((size_t)j << 12));
  for (int idx = tid; idx < 1024; idx += blockDim.x) dst[idx] = src[idx];
}

// ---------------------------------------------------------------------------
// gaussian_multi(carry, input) — input var is diagonal
// ---------------------------------------------------------------------------
__global__ void k_multi(int t, const int* __restrict__ sents,
                        const float* __restrict__ mu_emb,
                        const float* __restrict__ cho_emb,
                        const float* __restrict__ cs,
                        const float* __restrict__ cm,
                        const float* __restrict__ cv,
                        float* __restrict__ ms, float* __restrict__ mm,
                        float* __restrict__ mv) {
  __shared__ float sA[64 * LDP];
  __shared__ float sB[64 * LDP];
  __shared__ float sC[64 * LDP];
  __shared__ float ivd[64], imu[64], cmv[64], w0[64], w1[64], diff[64], red[64];
  __shared__ float sh[2];
  const int tid = threadIdx.x;
  const int b  = blockIdx.x >> 3;
  const int rem = blockIdx.x & 7;            // n = kk*ICOMP + ic
  const int kk = rem >> 1, ic = rem & 1;
  const int tok = sents[b * MAXLENV + t];
  const float* cvp = cv + ((size_t)(b * MAXKV + kk) << 12);
  const float* cmp = cm + (b * MAXKV + kk) * 64;

  if (tid < 64) {
    const float ch = cho_emb[tok * (ICOMPV * DIMV) + ic * 64 + tid];
    ivd[tid] = ch * ch;
    imu[tid] = mu_emb[tok * (ICOMPV * DIMV) + ic * 64 + tid];
    cmv[tid] = cmp[tid];
  }
  __syncthreads();
  for (int idx = tid; idx < 1024; idx += blockDim.x) {
    const int i = idx >> 4, j0 = (idx & 15) << 2;
    const float4 vv = ((const float4*)cvp)[idx];
    float* d = &sA[i * LDP + j0];
    d[0] = vv.x + (i == j0     ? ivd[i] : 0.f);
    d[1] = vv.y + (i == j0 + 1 ? ivd[i] : 0.f);
    d[2] = vv.z + (i == j0 + 2 ? ivd[i] : 0.f);
    d[3] = vv.w + (i == j0 + 3 ? ivd[i] : 0.f);
  }
  __syncthreads();
  chol64(sA);
  if (tid == 0) {
    float s = 0.f;
    for (int k = 0; k < 64; ++k) s += logf(sA[k * LDP + k]);
    sh[0] = 2.f * s;
  }
  __syncthreads();
  trinv64(sA, sB);
  wmma_mm(64, 64, 64, sB, LDP, true, sB, LDP, false, sC, LDP);  // inv(add)
  __syncthreads();
  if (tid < 64) {
    w0[tid] = dot64(sC + tid * LDP, cmv);   // inv @ mu0
    w1[tid] = dot64(sC + tid * LDP, imu);   // inv @ mu1
    diff[tid] = cmv[tid] - imu[tid];
  }
  __syncthreads();
  wmma_mm(64, 64, 64, cvp, 64, false, sC, LDP, false, sB, LDP); // var0 @ inv
  __syncthreads();
  float* mvp = mv + ((size_t)(b * 8 + rem) << 12);
  for (int idx = tid; idx < 1024; idx += blockDim.x) {
    const int i = idx >> 4, j0 = (idx & 15) << 2;
    const float* r = &sB[i * LDP + j0];
    float4 o;
    o.x = r[0] * ivd[j0];     o.y = r[1] * ivd[j0 + 1];
    o.z = r[2] * ivd[j0 + 2]; o.w = r[3] * ivd[j0 + 3];
    ((float4*)mvp)[idx] = o;                // (var0@inv)@diag(var1)
  }
  if (tid < 64) {
    mm[(b * 8 + rem) * 64 + tid] =
        ivd[tid] * w0[tid] + dot64(cvp + tid * 64, w1);
    red[tid] = diff[tid] * dot64(sC + tid * LDP, diff);
  }
  __syncthreads();
  if (tid == 0) {
    float m = 0.f;
    for (int i = 0; i < 64; ++i) m += red[i];
    const float zeta = -0.5f * (64.f * LOG2PI_F + sh[0] + m);
    ms[b * 8 + rem] = cs[b * MAXKV + kk] + zeta;
  }
}

// ---------------------------------------------------------------------------
// gaussian_multi_integral_fwd(t_mu, mid, t_var, mid_var)
// ---------------------------------------------------------------------------
__global__ void k_integral(const float* __restrict__ tmu,
                           const float* __restrict__ tv,
                           const float* __restrict__ ms,
                           const float* __restrict__ mm,
                           const float* __restrict__ mv,
                           float* __restrict__ qs, float* __restrict__ qm,
                           float* __restrict__ qv) {
  __shared__ float sA[64 * LDP];
  __shared__ float sB[64 * LDP];
  __shared__ float sC[64 * LDP];
  __shared__ float diff[64], red[64], sh[2];
  const int tid = threadIdx.x;
  const int id = blockIdx.x;
  const int c = id & 3, n = (id >> 2) & 7, b = id >> 5;
  const float* tvc = tv + c * TDV * TDV;
  const float* v1 = mv + ((size_t)(b * 8 + n) << 12);
  const int qi = b * 32 + n * 4 + c;

  if (tid < 64) diff[tid] = mm[(b * 8 + n) * 64 + tid] - tmu[c * TDV + tid];
  for (int idx = tid; idx < 1024; idx += blockDim.x) {   // S = Sxx + var1
    const int i = idx >> 4, j0 = (idx & 15) << 2;
    const float4 sx = *(const float4*)(tvc + i * TDV + j0);
    const float4 vv = ((const float4*)v1)[idx];
    float* d = &sA[i * LDP + j0];
    d[0] = sx.x + vv.x; d[1] = sx.y + vv.y;
    d[2] = sx.z + vv.z; d[3] = sx.w + vv.w;
  }
  __syncthreads();
  chol64(sA);
  if (tid == 0) {
    float s = 0.f;
    for (int k = 0; k < 64; ++k) s += logf(sA[k * LDP + k]);
    sh[0] = 2.f * s;
  }
  __syncthreads();
  trinv64(sA, sB);
  wmma_mm(64, 64, 64, sB, LDP, true, sB, LDP, false, sC, LDP);  // Sinv
  __syncthreads();
  wmma_mm(64, 64, 64, tvc + 64 * TDV, TDV, false, sC, LDP, false, sB, LDP); // K
  __syncthreads();
  if (tid < 64)
    qm[qi * 64 + tid] = tmu[c * TDV + 64 + tid] + dot64(sB + tid * LDP, diff);
  __syncthreads();
  wmma_mm(64, 64, 64, sB, LDP, false, tvc + 64, TDV, false, sA, LDP); // K@Sxy
  __syncthreads();
  for (int idx = tid; idx < 1024; idx += blockDim.x) {
    const int i = idx >> 4, j0 = (idx & 15) << 2;
    const float4 sy = *(const float4*)(tvc + (64 + i) * TDV + 64 + j0);
    const float* m2 = &sA[i * LDP + j0];
    float4 o;
    o.x = sy.x - m2[0]; o.y = sy.y - m2[1];
    o.z = sy.z - m2[2]; o.w = sy.w - m2[3];
    ((float4*)(qv + ((size_t)qi << 12)))[idx] = o;
  }
  if (tid < 64) red[tid] = diff[tid] * dot64(sC + tid * LDP, diff);
  __syncthreads();
  if (tid == 0) {
    float m = 0.f;
    for (int i = 0; i < 64; ++i) m += red[i];
    qs[qi] = ms[b * 8 + n] - 0.5f * (64.f * LOG2PI_F + sh[0] + m);
  }
}

// ---------------------------------------------------------------------------
// top-4 prune of 32 candidates per batch; snapshot carry when t==length-1
// ---------------------------------------------------------------------------
__global__ void k_topk(int t, const int* __restrict__ length,
                       const float* __restrict__ qs,
                       const float* __restrict__ qm,
                       const float* __restrict__ qv, float* __restrict__ cs,
                       float* __restrict__ cm, float* __restrict__ cv,
                       float* __restrict__ fs, float* __restrict__ fm,
                       float* __restrict__ fv) {
  __shared__ int sel[MAXKV];
  __shared__ float ssc[MAXKV];
  const int tid = threadIdx.x;
  const int b = blockIdx.x;
  if (tid == 0) {
    float sc[32];
    for (int j = 0; j < 32; ++j) sc[j] = qs[b * 32 + j];
    for (int k = 0; k < MAXKV; ++k) {
      float best = -INFINITY; int bi = 0;
      for (int j = 0; j < 32; ++j)
        if (sc[j] > best) { best = sc[j]; bi = j; }
      sel[k] = bi; ssc[k] = best; sc[bi] = -INFINITY;
    }
  }
  __syncthreads();
  for (int k = 0; k < MAXKV; ++k) {
    const int j = sel[k];
    if (tid == 0) cs[b * 4 + k] = ssc[k];
    if (tid < 64) cm[(b * 4 + k) * 64 + tid] = qm[(b * 32 + j) * 64 + tid];
    float4* dst = (float4*)(cv + ((size_t)(b * 4 + k) << 12));
    const float4* src = (const float4*)(qv + ((size_t)(b * 32 + j) << 12));
    for (int idx = tid; idx < 1024; idx += blockDim.x) dst[idx] = src[idx];
  }
  __syncthreads();
  if (t == length[b] - 1) {
    if (tid < 4) fs[b * 4 + tid] = cs[b * 4 + tid];
    float4* fmd = (float4*)(fm + b * 256);
    const float4* fms = (const float4*)(cm + b * 256);
    for (int idx = tid; idx < 64; idx += blockDim.x) fmd[idx] = fms[idx];
    float4* fvd = (float4*)(fv + (size_t)b * 16384);
    const float4* fvs = (const float4*)(cv + (size_t)b * 16384);
    for (int idx = tid; idx < 4096; idx += blockDim.x) fvd[idx] = fvs[idx];
  }
}

// ---------------------------------------------------------------------------
// final scoring: zeta of gaussian_multi(final, label) — only needs Cholesky
// ---------------------------------------------------------------------------
__global__ void k_score1(const float* __restrict__ omu,
                         const float* __restrict__ ocho,
                         const float* __restrict__ fs,
                         const float* __restrict__ fm,
                         const float* __restrict__ fv,
                         float* __restrict__ sc8) {
  __shared__ float sA[64 * LDP];
  __shared__ float diff[64], yv[64], red[64], sh[1];
  const int tid = threadIdx.x;   // blockDim == 64
  const int id = blockIdx.x;
  const int oc = id & 1, kk = (id >> 1) & 3, lab = (id >> 3) % NLABV,
            b = id / (NLABV * 8);
  const float ch = ocho[lab * (OCOMPV * DIMV) + oc * 64 + tid];
  const float ovd = ch * ch;
  diff[tid] = fm[(b * 4 + kk) * 64 + tid] - omu[lab * (OCOMPV * DIMV) + oc * 64 + tid];
  const float* fvp = fv + ((size_t)(b * 4 + kk) << 12);
  for (int j0 = 0; j0 < 64; j0 += 4) {
    const float4 vv = *(const float4*)(fvp + tid * 64 + j0);
    float* d = &sA[tid * LDP + j0];
    d[0] = vv.x; d[1] = vv.y; d[2] = vv.z; d[3] = vv.w;
  }
  sA[tid * LDP + tid] += ovd;
  __syncthreads();
  chol64(sA);
  if (tid == 0) {
    float s = 0.f;
    for (int k = 0; k < 64; ++k) s += logf(sA[k * LDP + k]);
    sh[0] = 2.f * s;
  }
  __syncthreads();
  float acc = 0.f;                      // forward solve L y = diff
  for (int k = 0; k < 64; ++k) {
    if (tid == k) yv[k] = (diff[k] - acc) / sA[k * LDP + k];
    __syncthreads();
    if (tid > k) acc += sA[tid * LDP + k] * yv[k];
    __syncthreads();
  }
  red[tid] = yv[tid] * yv[tid];
  __syncthreads();
  if (tid == 0) {
    float m = 0.f;
    for (int i = 0; i < 64; ++i) m += red[i];
    sc8[((b * NLABV + lab) * 8) + kk * 2 + oc] =
        fs[b * 4 + kk] - 0.5f * (64.f * LOG2PI_F + sh[0] + m);
  }
}

__global__ void k_score2(const float* __restrict__ sc8, float* __restrict__ out) {
  const int g = blockIdx.x * blockDim.x + threadIdx.x;
  if (g >= BATCHV * NLABV) return;
  float mx = -INFINITY;
  for (int j = 0; j < 8; ++j) mx = fmaxf(mx, sc8[g * 8 + j]);
  float s = 0.f;
  for (int j = 0; j < 8; ++j) s += expf(sc8[g * 8 + j] - mx);
  out[g] = mx + logf(s);
}

// ---------------------------------------------------------------------------
extern "C" void kernel_launch(void* const* d_in, const int* in_sizes, int n_in,
                              void* d_out, int out_size, void* d_ws,
                              size_t ws_size, hipStream_t stream) {
  (void)in_sizes; (void)n_in; (void)out_size; (void)ws_size;
  const int* sents      = (const int*)d_in[0];
  const int* length     = (const int*)d_in[1];
  const float* mu_emb   = (const float*)d_in[2];
  const float* cho_emb  = (const float*)d_in[3];
  const float* tmu      = (const float*)d_in[4];
  const float* tcho     = (const float*)d_in[5];
  const float* omu      = (const float*)d_in[6];
  const float* ocho     = (const float*)d_in[7];
  float* out = (float*)d_out;
  float* ws  = (float*)d_ws;

  float* tv  = ws + OFF_TVAR;
  float* is_ = ws + OFF_INITS; float* im_ = ws + OFF_INITM; float* iv_ = ws + OFF_INITV;
  float* cs  = ws + OFF_CS;    float* cm  = ws + OFF_CM;    float* cv  = ws + OFF_CV;
  float* ms  = ws + OFF_MS;    float* mm  = ws + OFF_MM;    float* mv  = ws + OFF_MV;
  float* qs  = ws + OFF_QS;    float* qm  = ws + OFF_QM;    float* qv  = ws + OFF_QV;
  float* fs  = ws + OFF_FS;    float* fm  = ws + OFF_FM;    float* fv  = ws + OFF_FV;
  float* sc8 = ws + OFF_SC8;

  k_tvar<<<TCOMPV, 256, 0, stream>>>(tcho, tv);
  k_init<<<TCOMPV, 256, 0, stream>>>(tmu, tv, is_, im_, iv_);
  k_init_scatter<<<BATCHV * MAXKV, 256, 0, stream>>>(is_, im_, iv_, cs, cm, cv);
  for (int t = 0; t < MAXLENV; ++t) {
    k_multi<<<BATCHV * MAXKV * ICOMPV, 256, 0, stream>>>(
        t, sents, mu_emb, cho_emb, cs, cm, cv, ms, mm, mv);
    k_integral<<<BATCHV * MAXKV * ICOMPV * TCOMPV, 256, 0, stream>>>(
        tmu, tv, ms, mm, mv, qs, qm, qv);
    k_topk<<<BATCHV, 256, 0, stream>>>(t, length, qs, qm, qv, cs, cm, cv, fs,
                                       fm, fv);
  }
  k_score1<<<BATCHV * NLABV * MAXKV * OCOMPV, 64, 0, stream>>>(omu, ocho, fs,
                                                               fm, fv, sc8);
  k_score2<<<(BATCHV * NLABV + 255) / 256, 256, 0, stream>>>(sc8, out);
}